// GAT2_429496729743
// MI455X (gfx1250) — compile-verified
//
#include <hip/hip_runtime.h>
#include <math.h>

typedef float v2f __attribute__((ext_vector_type(2)));
typedef float v8f __attribute__((ext_vector_type(8)));

#define NEG_SLOPE 0.2f

__device__ __forceinline__ float gat_lrelu(float v) { return v > 0.0f ? v : NEG_SLOPE * v; }
// monotone float->uint key so atomicMax(u32) implements float max (incl. negatives)
__device__ __forceinline__ unsigned gat_fkey(float f) {
    unsigned u = __float_as_uint(f);
    return (u & 0x80000000u) ? ~u : (u | 0x80000000u);
}
__device__ __forceinline__ float gat_funkey(unsigned k) {
    return __uint_as_float((k & 0x80000000u) ? (k & 0x7fffffffu) : ~k);
}

// ---------------------------------------------------------------------------
// Y[N, ncolStore] = X[N,128] @ W[128, ncolStore]   (fp32 WMMA 16x16x4)
// W staged in LDS, zero-padded to NCOL = NCT*16 columns.
// 256 threads = 8 waves; each wave computes a 16-row x NCOL output stripe.
// ---------------------------------------------------------------------------
template<int NCOL, int NCT>
__global__ __launch_bounds__(256) void gat_gemm_wmma(
    const float* __restrict__ X, const float* __restrict__ W,
    float* __restrict__ Y, int N, int ncolStore)
{
    constexpr int K = 128;
    __shared__ float ldsW[K * NCOL];
    for (int idx = threadIdx.x; idx < K * NCOL; idx += 256) {
        int k = idx / NCOL, c = idx % NCOL;
        ldsW[idx] = (c < ncolStore) ? W[k * ncolStore + c] : 0.0f;
    }
    __syncthreads();

    const int lane = threadIdx.x & 31;
    const int wave = threadIdx.x >> 5;
    const int m    = lane & 15;          // A row / B,C,D column within tile
    const int half = lane >> 4;          // 0: lanes 0-15, 1: lanes 16-31
    const int koff = half << 1;          // A/B K offset per ISA fragment layout

    long row0 = (long)blockIdx.x * 128 + wave * 16;
    long rowL = row0 + m; if (rowL > (long)N - 1) rowL = (long)N - 1;  // clamp (keep EXEC full)
    const float* xrow = X + rowL * K;

    v8f acc[NCT] = {};
    for (int k0 = 0; k0 < K; k0 += 4) {
        v2f a;
        a.x = xrow[k0 + koff];
        a.y = xrow[k0 + koff + 1];
#pragma unroll
        for (int t = 0; t < NCT; ++t) {
            v2f b;
            b.x = ldsW[(k0 + koff) * NCOL + t * 16 + m];
            b.y = ldsW[(k0 + koff + 1) * NCOL + t * 16 + m];
            acc[t] = __builtin_amdgcn_wmma_f32_16x16x4_f32(
                false, a, false, b, (short)0, acc[t], false, false);
        }
    }
    // C/D layout: VGPR r -> M = r + half*8, col = t*16 + (lane&15)
#pragma unroll
    for (int r = 0; r < 8; ++r) {
        long row = row0 + half * 8 + r;
        if (row < N) {
#pragma unroll
            for (int t = 0; t < NCT; ++t) {
                int c = t * 16 + m;
                if (c < ncolStore) Y[row * ncolStore + c] = acc[t][r];
            }
        }
    }
}

// ---------------------------------------------------------------------------
// alpha_s[n,h] = sum_c h[n,h,c]*a_src[h,c]; same for alpha_d. Wave per node.
// ---------------------------------------------------------------------------
template<int H, int C>
__global__ __launch_bounds__(256) void gat_alpha_kernel(
    const float* __restrict__ h, const float* __restrict__ aw_s,
    const float* __restrict__ aw_d, float* __restrict__ as_,
    float* __restrict__ ad_, int N)
{
    constexpr int HC = H * C;
    int wid  = blockIdx.x * (blockDim.x >> 5) + (threadIdx.x >> 5);
    int lane = threadIdx.x & 31;
    if (wid >= N) return;
    const float* hp = h + (long)wid * HC;
    float s[H], d[H];
#pragma unroll
    for (int hh = 0; hh < H; ++hh) { s[hh] = 0.0f; d[hh] = 0.0f; }
    for (int c = lane; c < HC; c += 32) {
        int hh = (H == 1) ? 0 : (c / C);
        float v = hp[c];
        s[hh] += v * aw_s[c];
        d[hh] += v * aw_d[c];
    }
#pragma unroll
    for (int hh = 0; hh < H; ++hh) {
        float sv = s[hh], dv = d[hh];
        for (int o = 16; o > 0; o >>= 1) {
            sv += __shfl_xor(sv, o, 32);
            dv += __shfl_xor(dv, o, 32);
        }
        if (lane == 0) {
            as_[(long)wid * H + hh] = sv;
            ad_[(long)wid * H + hh] = dv;
        }
    }
}

// ---------------------------------------------------------------------------
// Edge softmax passes (self-loops implicit: edge id >= E -> node (id-E))
// ---------------------------------------------------------------------------
template<int H>
__global__ __launch_bounds__(256) void gat_edge_max_kernel(
    const int* __restrict__ src, const int* __restrict__ dst, int E, int N,
    const float* __restrict__ as_, const float* __restrict__ ad_,
    unsigned* __restrict__ nmax)
{
    int e = blockIdx.x * blockDim.x + threadIdx.x;
    if (e >= E + N) return;
    int s = (e < E) ? src[e] : (e - E);
    int d = (e < E) ? dst[e] : (e - E);
#pragma unroll
    for (int h = 0; h < H; ++h) {
        float ev = gat_lrelu(as_[s * H + h] + ad_[d * H + h]);
        atomicMax(&nmax[d * H + h], gat_fkey(ev));
    }
}

template<int H>
__global__ __launch_bounds__(256) void gat_edge_denom_kernel(
    const int* __restrict__ src, const int* __restrict__ dst, int E, int N,
    const float* __restrict__ as_, const float* __restrict__ ad_,
    const unsigned* __restrict__ nmax, float* __restrict__ den)
{
    int e = blockIdx.x * blockDim.x + threadIdx.x;
    if (e >= E + N) return;
    int s = (e < E) ? src[e] : (e - E);
    int d = (e < E) ? dst[e] : (e - E);
#pragma unroll
    for (int h = 0; h < H; ++h) {
        float ev = gat_lrelu(as_[s * H + h] + ad_[d * H + h]);
        atomicAdd(&den[d * H + h], expf(ev - gat_funkey(nmax[d * H + h])));
    }
}

template<int H>
__global__ __launch_bounds__(256) void gat_edge_coef_kernel(
    const int* __restrict__ src, const int* __restrict__ dst, int E, int N,
    const float* __restrict__ as_, const float* __restrict__ ad_,
    const unsigned* __restrict__ nmax, const float* __restrict__ den,
    float* __restrict__ coef)
{
    int e = blockIdx.x * blockDim.x + threadIdx.x;
    if (e >= E + N) return;
    int s = (e < E) ? src[e] : (e - E);
    int d = (e < E) ? dst[e] : (e - E);
#pragma unroll
    for (int h = 0; h < H; ++h) {
        float ev = gat_lrelu(as_[s * H + h] + ad_[d * H + h]);
        float mx = gat_funkey(nmax[d * H + h]);
        coef[(long)e * H + h] = expf(ev - mx) / (den[d * H + h] + 1e-16f);
    }
}

// wave-per-edge scatter: agg[dst] += coef * feat[src]; coalesced rows
template<int H, int C>
__global__ __launch_bounds__(256) void gat_edge_scatter_kernel(
    const int* __restrict__ src, const int* __restrict__ dst, int E, int N,
    const float* __restrict__ coef, const float* __restrict__ feat,
    float* __restrict__ agg)
{
    constexpr int HC = H * C;
    int wid  = blockIdx.x * (blockDim.x >> 5) + (threadIdx.x >> 5);
    int lane = threadIdx.x & 31;
    if (wid >= E + N) return;
    int s = (wid < E) ? src[wid] : (wid - E);
    int d = (wid < E) ? dst[wid] : (wid - E);
    const float* fs = feat + (long)s * HC;
    float* ao = agg + (long)d * HC;
    for (int c = lane; c < HC; c += 32) {
        int hh = (H == 1) ? 0 : (c / C);
        atomicAdd(&ao[c], coef[(long)wid * H + hh] * fs[c]);
    }
}

// ---------------------------------------------------------------------------
__global__ void gat_fill_u32_kernel(unsigned* __restrict__ p, unsigned v, long n) {
    long i = blockIdx.x * (long)blockDim.x + threadIdx.x;
    if (i < n) p[i] = v;
}

__global__ void gat_elu_bias_kernel(float* __restrict__ a, const float* __restrict__ b,
                                    long n, int HC) {
    long i = blockIdx.x * (long)blockDim.x + threadIdx.x;
    if (i < n) {
        float t = a[i] + b[i % HC];
        a[i] = t > 0.0f ? t : (expf(t) - 1.0f);
    }
}

__global__ void gat_bias_out_kernel(const float* __restrict__ a, const float* __restrict__ b,
                                    float* __restrict__ out, long n, int HC) {
    long i = blockIdx.x * (long)blockDim.x + threadIdx.x;
    if (i < n) out[i] = a[i] + b[i % HC];
}

// ---------------------------------------------------------------------------
extern "C" void kernel_launch(void* const* d_in, const int* in_sizes, int n_in,
                              void* d_out, int out_size, void* d_ws, size_t ws_size,
                              hipStream_t stream)
{
    (void)n_in; (void)out_size; (void)ws_size;
    const float* x   = (const float*)d_in[0];
    const int*   ei  = (const int*)  d_in[1];
    const float* W1  = (const float*)d_in[2];
    const float* aS1 = (const float*)d_in[3];
    const float* aD1 = (const float*)d_in[4];
    const float* b1  = (const float*)d_in[5];
    const float* W2  = (const float*)d_in[6];
    const float* aS2 = (const float*)d_in[7];
    const float* aD2 = (const float*)d_in[8];
    const float* b2  = (const float*)d_in[9];

    const int IN = 128, HC1 = 128, OUTC = 40;
    int N  = in_sizes[0] / IN;
    int E  = in_sizes[1] / 2;
    int ET = E + N;
    const int* srcI = ei;
    const int* dstI = ei + E;

    float* wsf = (float*)d_ws;
    size_t off = 0;
    auto take = [&](size_t nf) -> size_t { size_t o = off; off += (nf + 63) & ~(size_t)63; return o; };
    float*    h1    = wsf + take((size_t)N * HC1);
    float*    agg1  = wsf + take((size_t)N * HC1);
    float*    as1   = wsf + take((size_t)N * 2);
    float*    ad1   = wsf + take((size_t)N * 2);
    unsigned* nmax1 = (unsigned*)(wsf + take((size_t)N * 2));
    float*    den1  = wsf + take((size_t)N * 2);
    float*    coef1 = wsf + take((size_t)ET * 2);
    float*    h2    = wsf + take((size_t)N * OUTC);
    float*    agg2  = wsf + take((size_t)N * OUTC);
    float*    as2   = wsf + take((size_t)N);
    float*    ad2   = wsf + take((size_t)N);
    unsigned* nmax2 = (unsigned*)(wsf + take((size_t)N));
    float*    den2  = wsf + take((size_t)N);
    float*    coef2 = wsf + take((size_t)ET);

    dim3 blk(256);
    int egrid = (ET + 255) / 256;
    int wgrid = (ET + 7) / 8;        // wave-per-edge

    // ---------------- layer 1 (H=2, C=64, concat) ----------------
    gat_gemm_wmma<128, 8><<<(N + 127) / 128, blk, 0, stream>>>(x, W1, h1, N, HC1);
    gat_alpha_kernel<2, 64><<<(N + 7) / 8, blk, 0, stream>>>(h1, aS1, aD1, as1, ad1, N);
    gat_fill_u32_kernel<<<((long)N * 2 + 255) / 256, blk, 0, stream>>>(nmax1, 0u, (long)N * 2);
    gat_fill_u32_kernel<<<((long)N * 2 + 255) / 256, blk, 0, stream>>>((unsigned*)den1, 0u, (long)N * 2);
    gat_fill_u32_kernel<<<((long)N * HC1 + 255) / 256, blk, 0, stream>>>((unsigned*)agg1, 0u, (long)N * HC1);
    gat_edge_max_kernel<2><<<egrid, blk, 0, stream>>>(srcI, dstI, E, N, as1, ad1, nmax1);
    gat_edge_denom_kernel<2><<<egrid, blk, 0, stream>>>(srcI, dstI, E, N, as1, ad1, nmax1, den1);
    gat_edge_coef_kernel<2><<<egrid, blk, 0, stream>>>(srcI, dstI, E, N, as1, ad1, nmax1, den1, coef1);
    gat_edge_scatter_kernel<2, 64><<<wgrid, blk, 0, stream>>>(srcI, dstI, E, N, coef1, h1, agg1);
    gat_elu_bias_kernel<<<((long)N * HC1 + 255) / 256, blk, 0, stream>>>(agg1, b1, (long)N * HC1, HC1);

    // ---------------- layer 2 (H=1, C=40, mean==identity) ----------------
    gat_gemm_wmma<48, 3><<<(N + 127) / 128, blk, 0, stream>>>(agg1, W2, h2, N, OUTC);
    gat_alpha_kernel<1, 40><<<(N + 7) / 8, blk, 0, stream>>>(h2, aS2, aD2, as2, ad2, N);
    gat_fill_u32_kernel<<<((long)N + 255) / 256, blk, 0, stream>>>(nmax2, 0u, (long)N);
    gat_fill_u32_kernel<<<((long)N + 255) / 256, blk, 0, stream>>>((unsigned*)den2, 0u, (long)N);
    gat_fill_u32_kernel<<<((long)N * OUTC + 255) / 256, blk, 0, stream>>>((unsigned*)agg2, 0u, (long)N * OUTC);
    gat_edge_max_kernel<1><<<egrid, blk, 0, stream>>>(srcI, dstI, E, N, as2, ad2, nmax2);
    gat_edge_denom_kernel<1><<<egrid, blk, 0, stream>>>(srcI, dstI, E, N, as2, ad2, nmax2, den2);
    gat_edge_coef_kernel<1><<<egrid, blk, 0, stream>>>(srcI, dstI, E, N, as2, ad2, nmax2, den2, coef2);
    gat_edge_scatter_kernel<1, 40><<<wgrid, blk, 0, stream>>>(srcI, dstI, E, N, coef2, h2, agg2);
    gat_bias_out_kernel<<<((long)N * OUTC + 255) / 256, blk, 0, stream>>>(
        agg2, b2, (float*)d_out, (long)N * OUTC, OUTC);
}